// QuantumLayer_84713934946889
// MI455X (gfx1250) — compile-verified
//
#include <hip/hip_runtime.h>
#include <hip/hip_bf16.h>

// ---------------------------------------------------------------------------
// 4-qubit, 4-layer StronglyEntangling circuit over B samples.
//
// Math restructure: layers are batch-independent => fold them into one fixed
// 16x16 complex unitary U (built on-device from q_weights, 2KB in d_ws).
// The RY embedding of |0000> is a REAL product vector v (16 floats/sample).
// Per 16-sample tile:  Y_R = Re(U) * V,  Y_I = Im(U) * V   (V is 16x16 f32)
// via V_WMMA_F32_16X16X4_F32 chained over K (4 chunks of K=4), then
// p_i = yR_i^2 + yI_i^2 and <Z_w> = sum_i sign_w(i) * p_i.
//
// WMMA operand mapping (wave32, per CDNA5 ISA 7.12.2):
//   A (16x4 f32, 2 VGPRs): lane<16 -> row M=lane, K=0,1 ; lane>=16 -> M=lane-16, K=2,3
//   B (4x16 f32, 2 VGPRs): lane<16 -> col N=lane, rows K=0,1 ; lane>=16 -> rows K=2,3
//   C/D (16x16, 8 VGPRs):  lane<16 -> (M=r, N=lane) ; lane>=16 -> (M=r+8, N=lane-16)
// Basis index i = 8*b0 + 4*b1 + 2*b2 + b3 (qubit w <-> bit 3-w).
// Chunk kc covers rows 4kc..4kc+3: b0b1 = kc, b2 = lane-half, b3 = VGPR slot.
// ---------------------------------------------------------------------------

typedef float v2f __attribute__((ext_vector_type(2)));
typedef float v4f __attribute__((ext_vector_type(4)));
typedef float v8f __attribute__((ext_vector_type(8)));

#define NQ 4
#define NLAYERS 4

// ---------------------------------------------------------------------------
// Kernel 1: build U = (layer gates, applied in program order) as 16x16 complex.
// Thread t < 16 owns column t (columns transform independently under G*U).
// Everything fully unrolled -> columns stay in VGPRs.
// ---------------------------------------------------------------------------
__global__ __launch_bounds__(32) void build_unitary_kernel(
    const float* __restrict__ qw, float* __restrict__ Ur, float* __restrict__ Ui) {
  const int t = threadIdx.x;
  if (t >= 16) return;

  float cr[16], ci[16];
#pragma unroll
  for (int i = 0; i < 16; ++i) { cr[i] = (i == t) ? 1.0f : 0.0f; ci[i] = 0.0f; }

  for (int l = 0; l < NLAYERS; ++l) {
    // Rot(phi, theta, omega) on each wire:
    //   u00 = em*c, u01 = -ed*s, u10 = conj(ed)*s, u11 = conj(em)*c
    //   em = e^{-i(phi+omega)/2}, ed = e^{+i(phi-omega)/2}
#pragma unroll
    for (int w = 0; w < NQ; ++w) {
      const float phi = qw[(l * NQ + w) * 3 + 0];
      const float th  = qw[(l * NQ + w) * 3 + 1];
      const float om  = qw[(l * NQ + w) * 3 + 2];
      float sh, ch;  __sincosf(0.5f * th, &sh, &ch);
      float emi, emr; __sincosf(-0.5f * (phi + om), &emi, &emr);
      float edi, edr; __sincosf( 0.5f * (phi - om), &edi, &edr);
      const float u00r =  emr * ch, u00i =  emi * ch;
      const float u01r = -edr * sh, u01i = -edi * sh;
      const float u10r =  edr * sh, u10i = -edi * sh;
      const float u11r =  emr * ch, u11i = -emi * ch;
      const int mask = 8 >> w;  // wire w <-> bit (3-w)
#pragma unroll
      for (int i = 0; i < 16; ++i) {
        if (i & mask) continue;           // compile-time folded
        const int j = i | mask;
        const float a0r = cr[i], a0i = ci[i], a1r = cr[j], a1i = ci[j];
        cr[i] = u00r * a0r - u00i * a0i + u01r * a1r - u01i * a1i;
        ci[i] = u00r * a0i + u00i * a0r + u01r * a1i + u01i * a1r;
        cr[j] = u10r * a0r - u10i * a0i + u11r * a1r - u11i * a1i;
        ci[j] = u10r * a0i + u10i * a0r + u11r * a1i + u11i * a1r;
      }
    }
    // CNOT ring (0,1),(1,2),(2,3),(3,0): pure amplitude permutation.
#pragma unroll
    for (int w = 0; w < NQ; ++w) {
      const int cbit = 8 >> w;
      const int tbit = 8 >> ((w + 1) & 3);
#pragma unroll
      for (int i = 0; i < 16; ++i) {
        if ((i & cbit) && !(i & tbit)) {
          const int j = i | tbit;
          float tr = cr[i]; cr[i] = cr[j]; cr[j] = tr;
          float ti = ci[i]; ci[i] = ci[j]; ci[j] = ti;
        }
      }
    }
  }
#pragma unroll
  for (int i = 0; i < 16; ++i) { Ur[i * 16 + t] = cr[i]; Ui[i * 16 + t] = ci[i]; }
}

// ---------------------------------------------------------------------------
// Kernel 2: batched simulation. 1 wave handles 16 samples/tile, 8 WMMAs/tile.
// ---------------------------------------------------------------------------
__global__ __launch_bounds__(256) void qsim_wmma_kernel(
    const v4f* __restrict__ in, const float* __restrict__ Ur,
    const float* __restrict__ Ui, v4f* __restrict__ out,
    int nSamples, int tilesPerWave) {
  const int lane   = threadIdx.x & 31;
  const int waveId = blockIdx.x * (blockDim.x >> 5) + (threadIdx.x >> 5);
  const int h = lane >> 4;    // lane half: selects K rows / output-row half
  const int m = lane & 15;    // A-matrix row M, and sample-in-tile N

  // A operands (constant over tiles): lane half h loads cols {4kc+2h, 4kc+2h+1}.
  v2f ar[4], ai[4];
  const int cb = h << 1;
#pragma unroll
  for (int kc = 0; kc < 4; ++kc) {
    ar[kc] = *(const v2f*)(Ur + m * 16 + 4 * kc + cb);
    ai[kc] = *(const v2f*)(Ui + m * 16 + 4 * kc + cb);
  }

  const int waveBase = waveId * (tilesPerWave << 4);
  for (int t = 0; t < tilesPerWave; ++t) {
    const int tileBase = waveBase + (t << 4);
    if (tileBase + 16 > nSamples) break;  // uniform per wave: EXEC stays all-1s
    const int s = tileBase + m;           // lanes L and L+16 share sample s

    const v4f x = in[s];
    float c0, s0, c1, s1, c2, s2, c3, s3;
    __sincosf(0.5f * x.x, &s0, &c0);
    __sincosf(0.5f * x.y, &s1, &c1);
    __sincosf(0.5f * x.z, &s2, &c2);
    __sincosf(0.5f * x.w, &s3, &c3);

    // v[i] = f0(b0)*f1(b1)*f2(b2)*f3(b3); this lane supplies rows with b2=h.
    const float f2  = h ? s2 : c2;
    const float f20 = f2 * c3;            // b3 = 0  (VGPR slot 0)
    const float f21 = f2 * s3;            // b3 = 1  (VGPR slot 1)
    const float f01[4] = { c0 * c1, c0 * s1, s0 * c1, s0 * s1 };  // b0b1 = kc

    v8f accR = {}; v8f accI = {};
#pragma unroll
    for (int kc = 0; kc < 4; ++kc) {
      v2f b; b.x = f01[kc] * f20; b.y = f01[kc] * f21;
      accR = __builtin_amdgcn_wmma_f32_16x16x4_f32(
          false, ar[kc], false, b, (short)0, accR, false, false);
      accI = __builtin_amdgcn_wmma_f32_16x16x4_f32(
          false, ai[kc], false, b, (short)0, accI, false, false);
    }

    // p_r = |amp_{8h+r}|^2 of sample m.  i = 8h + r: b0=h, b1=r>>2, b2=(r>>1)&1, b3=r&1.
    float p[8];
#pragma unroll
    for (int r = 0; r < 8; ++r) p[r] = accR[r] * accR[r] + accI[r] * accI[r];

    const float lo = p[0] + p[1] + p[2] + p[3];
    const float hi = p[4] + p[5] + p[6] + p[7];
    float e0 = lo + hi;                                       // sign from b0 = h below
    float e1 = lo - hi;                                       // sign from b1
    float e2 = (p[0] + p[1] - p[2] - p[3]) + (p[4] + p[5] - p[6] - p[7]);  // b2
    float e3 = (p[0] - p[1] + p[2] - p[3]) + (p[4] - p[5] + p[6] - p[7]);  // b3
    if (h) e0 = -e0;

    // Combine the two halves (rows 0-7 vs 8-15 of the same sample).
    e0 += __shfl_xor(e0, 16, 32);
    e1 += __shfl_xor(e1, 16, 32);
    e2 += __shfl_xor(e2, 16, 32);
    e3 += __shfl_xor(e3, 16, 32);

    if (h == 0) {
      v4f o; o.x = e0; o.y = e1; o.z = e2; o.w = e3;
      __builtin_nontemporal_store(o, &out[s]);   // streaming: bypass cache rinse
    }
  }
}

// ---------------------------------------------------------------------------
extern "C" void kernel_launch(void* const* d_in, const int* in_sizes, int n_in,
                              void* d_out, int out_size, void* d_ws, size_t ws_size,
                              hipStream_t stream) {
  const float* inputs   = (const float*)d_in[0];   // (B, 4) f32
  const float* qweights = (const float*)d_in[1];   // (4, 4, 3) f32
  float* Ur = (float*)d_ws;        // 256 floats
  float* Ui = Ur + 256;            // 256 floats   (512 floats = 2KB of d_ws)

  const int nSamples = in_sizes[0] / 4;

  build_unitary_kernel<<<1, 32, 0, stream>>>(qweights, Ur, Ui);

  const int TPW = 8;                                // tiles per wave
  const int samplesPerWave  = 16 * TPW;             // 128
  const int wavesPerBlock   = 8;                    // 256 threads
  const int waves  = (nSamples + samplesPerWave - 1) / samplesPerWave;
  const int blocks = (waves + wavesPerBlock - 1) / wavesPerBlock;

  qsim_wmma_kernel<<<blocks, 256, 0, stream>>>(
      (const v4f*)inputs, Ur, Ui, (v4f*)d_out, nSamples, TPW);
}